// SimpleGNN_70523363000622
// MI455X (gfx1250) — compile-verified
//
#include <hip/hip_runtime.h>
#include <hip/hip_bf16.h>

typedef __attribute__((ext_vector_type(2))) float v2f;
typedef __attribute__((ext_vector_type(8))) float v8f;

#define N_NODES  20000
#define N_EDGES  320000
#define N_GRAPHS 64

// ---------------------------------------------------------------------------
// FP32 WMMA GEMM: C[M,N] = A[M,K] * B[K,N]   (row-major, no bias)
// Block = 8 waves. blockIdx.y selects a 32-column B panel, staged into LDS
// with GLOBAL_LOAD_ASYNC_TO_LDS_B128 (CDNA5 async data mover, ASYNCcnt) and
// shared by all waves; each wave computes a 32x32 tile (2x2 accumulators)
// via V_WMMA_F32_16X16X4_F32.  M % 32 == 0, N % 32 == 0, K % 8 == 0.
// LDS B panel rows padded to 40 floats: 2*40 = 64+16, so half-waves reading
// rows k/k+2 hit disjoint bank groups (conflict-free ds_load).
// ---------------------------------------------------------------------------
__global__ __launch_bounds__(256)
void gemm_wmma_f32(const float* __restrict__ A, const float* __restrict__ B,
                   float* __restrict__ C, int M, int N, int K) {
    __shared__ float bpanel[256 * 40];          // K <= 256, padded row stride 40
    const int tid   = threadIdx.x;
    const int lane  = tid & 31;
    const int wave  = tid >> 5;
    const int ncol0 = blockIdx.y << 5;          // 32-col panel base

    // ---- async-stage B panel [K x 32] into LDS: 16B chunks, cache -> LDS ----
    // chunk ch: row k = ch>>3, 4-float group c4 = ch&7 (rows are 32 floats, so
    // chunks never straddle the padded 40-float LDS row).
    for (int ch = tid; ch < (K << 3); ch += 256) {
        const int k  = ch >> 3;
        const int c4 = ch & 7;
        unsigned lds_off = (unsigned)(size_t)(&bpanel[k * 40 + (c4 << 2)]);
        unsigned long long gaddr =
            (unsigned long long)(size_t)(B + (long)k * N + ncol0 + (c4 << 2));
        asm volatile("global_load_async_to_lds_b128 %0, %1, off"
                     :: "v"(lds_off), "v"(gaddr) : "memory");
    }
    asm volatile("s_wait_asynccnt 0x0" ::: "memory");
    __syncthreads();

    const int mt = blockIdx.x * 8 + wave;       // 32-row group index
    if (mt < (M >> 5)) {                        // wave-uniform: EXEC all-1 inside
        const int l15   = lane & 15;
        const int khalf = (lane >> 4) << 1;     // 0 or 2 (K-pair of this half)
        const int row0  = (mt << 5) + l15;
        const float* a0p = A + (long)row0 * K + khalf;
        const float* a1p = a0p + ((long)K << 4);        // +16 rows
        const float* bl0 = bpanel + khalf * 40 + l15;

        v8f acc00 = {}, acc01 = {}, acc10 = {}, acc11 = {};
        #pragma unroll 2
        for (int k = 0; k < K; k += 4) {
            v2f a0, a1, b0, b1;
            a0.x = a0p[k];  a0.y = a0p[k + 1];
            a1.x = a1p[k];  a1.y = a1p[k + 1];
            const float* bk = bl0 + k * 40;
            b0.x = bk[0];   b0.y = bk[40];      // cols l15,     rows k+kh, k+kh+1
            b1.x = bk[16];  b1.y = bk[56];      // cols l15+16
            acc00 = __builtin_amdgcn_wmma_f32_16x16x4_f32(false, a0, false, b0,
                                                          (short)0, acc00, false, false);
            acc01 = __builtin_amdgcn_wmma_f32_16x16x4_f32(false, a0, false, b1,
                                                          (short)0, acc01, false, false);
            acc10 = __builtin_amdgcn_wmma_f32_16x16x4_f32(false, a1, false, b0,
                                                          (short)0, acc10, false, false);
            acc11 = __builtin_amdgcn_wmma_f32_16x16x4_f32(false, a1, false, b1,
                                                          (short)0, acc11, false, false);
        }

        // C/D layout: VGPR v holds row base + 8*(lane>>4) + v, col = l15
        const int rbase = (mt << 5) + (khalf << 2);
        float* cp = C + (long)rbase * N + ncol0 + l15;
        #pragma unroll
        for (int v = 0; v < 8; ++v) {
            cp[(long)v * N]             = acc00[v];
            cp[(long)v * N + 16]        = acc01[v];
            cp[(long)(v + 16) * N]      = acc10[v];
            cp[(long)(v + 16) * N + 16] = acc11[v];
        }
    }
}

// ---------------------------------------------------------------------------
// Utility / graph kernels
// ---------------------------------------------------------------------------
__global__ void fill_f32(float* __restrict__ p, float v, long n) {
    long i = (long)blockIdx.x * blockDim.x + threadIdx.x;
    if (i < n) p[i] = v;
}

__global__ void deg_count(const int* __restrict__ dst, float* __restrict__ deg, int E) {
    int e = blockIdx.x * blockDim.x + threadIdx.x;
    if (e < E) atomicAdd(&deg[dst[e]], 1.0f);
}

__global__ void make_dinv(const float* __restrict__ deg, float* __restrict__ dinv, int n) {
    int i = blockIdx.x * blockDim.x + threadIdx.x;
    if (i < n) dinv[i] = rsqrtf(deg[i]);
}

// agg[dst] += h[src] * dinv[src]*dinv[dst] ; one lane = 4 contiguous features
__global__ __launch_bounds__(256)
void edge_scatter(const float* __restrict__ h, const int* __restrict__ src,
                  const int* __restrict__ dst, const float* __restrict__ dinv,
                  float* __restrict__ agg, int E, int cshift /*log2(F/4)*/, int F) {
    long gid = (long)blockIdx.x * blockDim.x + threadIdx.x;
    long total = (long)E << cshift;
    if (gid >= total) return;
    int e = (int)(gid >> cshift);
    int c = (int)(gid & ((1 << cshift) - 1));
    int s = src[e], d = dst[e];
    float nrm = dinv[s] * dinv[d];
    float4 hv = ((const float4*)(h + (long)s * F))[c];
    float* op = agg + (long)d * F + (c << 2);
    atomicAdd(op + 0, hv.x * nrm);
    atomicAdd(op + 1, hv.y * nrm);
    atomicAdd(op + 2, hv.z * nrm);
    atomicAdd(op + 3, hv.w * nrm);
}

// agg = [relu](agg + h * dinv^2 + bias)
__global__ void epilogue(float* __restrict__ agg, const float* __restrict__ h,
                         const float* __restrict__ dinv, const float* __restrict__ bias,
                         long n, int fmask, int fshift /*log2 F*/, int relu) {
    long i = (long)blockIdx.x * blockDim.x + threadIdx.x;
    if (i >= n) return;
    int f    = (int)(i & fmask);
    int node = (int)(i >> fshift);
    float di = dinv[node];
    float v  = agg[i] + h[i] * di * di + bias[f];
    agg[i] = relu ? fmaxf(v, 0.0f) : v;
}

// mean-pool accumulation: out[g][f] += h[node][f]; cnt[g] += 1 (once per node)
__global__ void pool_sum(const float* __restrict__ h, const int* __restrict__ batch,
                         float* __restrict__ out, float* __restrict__ cnt) {
    long gid = (long)blockIdx.x * blockDim.x + threadIdx.x;
    if (gid >= (long)N_NODES * 64) return;
    int f    = (int)(gid & 63);
    int node = (int)(gid >> 6);
    int g    = batch[node];
    atomicAdd(&out[(long)g * 64 + f], h[gid]);
    if (f == 0) atomicAdd(&cnt[g], 1.0f);
}

__global__ void pool_div(float* __restrict__ out, const float* __restrict__ cnt) {
    int i = blockIdx.x * blockDim.x + threadIdx.x;
    if (i < N_GRAPHS * 64) out[i] /= fmaxf(cnt[i >> 6], 1.0f);
}

// ---------------------------------------------------------------------------
static inline int gridFor(long n, int b) { return (int)((n + b - 1) / b); }

extern "C" void kernel_launch(void* const* d_in, const int* in_sizes, int n_in,
                              void* d_out, int out_size, void* d_ws, size_t ws_size,
                              hipStream_t stream) {
    const float* x     = (const float*)d_in[0];
    const int*   ei    = (const int*)d_in[1];
    const int*   src   = ei;
    const int*   dst   = ei + N_EDGES;
    const int*   batch = (const int*)d_in[2];
    const float* W1 = (const float*)d_in[3]; const float* b1 = (const float*)d_in[4];
    const float* W2 = (const float*)d_in[5]; const float* b2 = (const float*)d_in[6];
    const float* W3 = (const float*)d_in[7]; const float* b3 = (const float*)d_in[8];
    float* out = (float*)d_out;

    // workspace layout (fp32): h[20000*256] | agg[20000*256] | deg | dinv | cnt
    float* hbuf = (float*)d_ws;
    float* agg  = hbuf + (long)N_NODES * 256;
    float* deg  = agg  + (long)N_NODES * 256;
    float* dinv = deg  + N_NODES;
    float* cnt  = dinv + N_NODES;

    const int B = 256;
    const int mBlocks = gridFor(N_NODES / 32, 8);   // 8 waves x 32 rows per block

    // ---- degrees (A + I, in-degree) -> dinv = rsqrt(deg) ----
    fill_f32<<<gridFor(N_NODES, B), B, 0, stream>>>(deg, 1.0f, N_NODES);
    deg_count<<<gridFor(N_EDGES, B), B, 0, stream>>>(dst, deg, N_EDGES);
    make_dinv<<<gridFor(N_NODES, B), B, 0, stream>>>(deg, dinv, N_NODES);

    // ---- layer 1: h = x @ W1 (128 -> 256), aggregate, +b1, relu ----
    {
        gemm_wmma_f32<<<dim3(mBlocks, 256 / 32), B, 0, stream>>>(x, W1, hbuf,
                                                                 N_NODES, 256, 128);
        long nAgg = (long)N_NODES * 256;
        fill_f32<<<gridFor(nAgg, B), B, 0, stream>>>(agg, 0.0f, nAgg);
        edge_scatter<<<gridFor((long)N_EDGES << 6, B), B, 0, stream>>>(
            hbuf, src, dst, dinv, agg, N_EDGES, 6, 256);
        epilogue<<<gridFor(nAgg, B), B, 0, stream>>>(agg, hbuf, dinv, b1, nAgg, 255, 8, 1);
    }

    // ---- layer 2: h = agg @ W2 (256 -> 256), aggregate, +b2, relu ----
    {
        gemm_wmma_f32<<<dim3(mBlocks, 256 / 32), B, 0, stream>>>(agg, W2, hbuf,
                                                                 N_NODES, 256, 256);
        long nAgg = (long)N_NODES * 256;
        fill_f32<<<gridFor(nAgg, B), B, 0, stream>>>(agg, 0.0f, nAgg);
        edge_scatter<<<gridFor((long)N_EDGES << 6, B), B, 0, stream>>>(
            hbuf, src, dst, dinv, agg, N_EDGES, 6, 256);
        epilogue<<<gridFor(nAgg, B), B, 0, stream>>>(agg, hbuf, dinv, b2, nAgg, 255, 8, 1);
    }

    // ---- layer 3: h = agg @ W3 (256 -> 64), aggregate, +b3 (no relu) ----
    {
        gemm_wmma_f32<<<dim3(mBlocks, 64 / 32), B, 0, stream>>>(agg, W3, hbuf,
                                                                N_NODES, 64, 256);
        long nAgg = (long)N_NODES * 64;
        fill_f32<<<gridFor(nAgg, B), B, 0, stream>>>(agg, 0.0f, nAgg);
        edge_scatter<<<gridFor((long)N_EDGES << 4, B), B, 0, stream>>>(
            hbuf, src, dst, dinv, agg, N_EDGES, 4, 64);
        epilogue<<<gridFor(nAgg, B), B, 0, stream>>>(agg, hbuf, dinv, b3, nAgg, 63, 6, 0);
    }

    // ---- global mean pool over 64 graphs -> out[64,64] ----
    fill_f32<<<gridFor(N_GRAPHS * 64, B), B, 0, stream>>>(out, 0.0f, N_GRAPHS * 64);
    fill_f32<<<gridFor(N_GRAPHS, B), B, 0, stream>>>(cnt, 0.0f, N_GRAPHS);
    pool_sum<<<gridFor((long)N_NODES * 64, B), B, 0, stream>>>(agg, batch, out, cnt);
    pool_div<<<gridFor(N_GRAPHS * 64, B), B, 0, stream>>>(out, cnt);
}